// CPhaseLayer_27470610825417
// MI455X (gfx1250) — compile-verified
//
#include <hip/hip_runtime.h>
#include <stdint.h>

// ---------------------------------------------------------------------------
// CPhase layer on MI455X (gfx1250).
//
// tmat = kron(CPHASE,...,CPHASE, I2) with CPHASE = diag(1,1,-1,1)  =>  tmat is
// DIAGONAL with entries +-1. out[b,c,m] = x[b,c,m] * diag[m]. This is a pure
// streaming op: 512 MiB of HBM traffic => ~23us at 23.3 TB/s. A dense WMMA
// GEMM would be ~1.1e12 FLOP (hundreds of us, and lossy), so matrix ops are
// deliberately not used; the optimal mix is b128 NT vmem traffic.
// ---------------------------------------------------------------------------

typedef float    v4f __attribute__((ext_vector_type(4)));
typedef uint32_t v4u __attribute__((ext_vector_type(4)));

#define NSTATES      8192                 // 2^13
#define VEC_PER_ROW  (NSTATES / 4)        // 2048 float4 per state vector
#define VEC_ROW_LOG2 11                   // log2(VEC_PER_ROW)

// Kernel 1: pull the diagonal of tmat (ground-truth signs) into workspace.
// 32 KB table -> cache resident for the whole main pass.
__global__ void cphase_extract_diag(const float* __restrict__ tmat,
                                    float* __restrict__ diag) {
    int m = blockIdx.x * blockDim.x + threadIdx.x;
    if (m < NSTATES) diag[m] = tmat[(size_t)m * (size_t)(NSTATES + 1)];
}

// Kernel 2: out[row, col] = x[row, col] * diag[col] over vec4 columns.
// Each thread owns ONE column (diag load hoisted out of the loop) and strides
// over rows. Consecutive lanes -> consecutive columns -> coalesced b128.
// NT hints on the big streams so they don't thrash L2; diag stays RT/cached.
__global__ void cphase_diag_scale(const float* __restrict__ x,
                                  const float* __restrict__ diag,
                                  float* __restrict__ out,
                                  long nrows) {
    const v4f* __restrict__ xv = (const v4f*)x;
    const v4f* __restrict__ dv = (const v4f*)diag;
    v4f* __restrict__ ov       = (v4f*)out;

    const long tid        = (long)blockIdx.x * (long)blockDim.x + threadIdx.x;
    const long nthreads   = (long)gridDim.x * (long)blockDim.x;
    const int  col        = (int)(tid & (VEC_PER_ROW - 1));
    const long row_stride = nthreads >> VEC_ROW_LOG2;   // exact: nthreads % 2048 == 0

    const v4f d = dv[col];                              // loop-invariant, cached

    for (long row = tid >> VEC_ROW_LOG2; row < nrows; row += row_stride) {
        const long i = (row << VEC_ROW_LOG2) + col;
        // CDNA5 prefetch path (global_prefetch_b8): next row-stride iteration.
        if (row + row_stride < nrows)
            __builtin_prefetch((const void*)&xv[i + (row_stride << VEC_ROW_LOG2)], 0, 0);

        v4f xval = __builtin_nontemporal_load(&xv[i]);  // global_load_b128 th:NT
        __builtin_nontemporal_store(xval * d, &ov[i]);  // global_store_b128 th:NT
    }
}

// Fallback (no workspace / odd shapes): signs computed analytically.
// Pair k occupies bits (12-2k, 11-2k); diag entry is -1 iff pair value == 2
// (high bit set, low bit clear). High-bit positions {12,10,8,6,4,2} = 0x1554.
// Bit 0 (the I2 qubit) never affects the sign; m%4==0 => bit1==0, and m+2
// makes pair(2,1) odd-valued (never 2), so s23 = s01 without the bit-2 term.
__global__ void cphase_sign_flip(const float* __restrict__ x,
                                 float* __restrict__ out,
                                 long total_vec) {
    const v4u* __restrict__ xv = (const v4u*)x;
    v4u* __restrict__ ov       = (v4u*)out;

    const long stride = (long)gridDim.x * (long)blockDim.x;
    for (long i = (long)blockIdx.x * (long)blockDim.x + threadIdx.x;
         i < total_vec; i += stride) {
        if (i + stride < total_vec)
            __builtin_prefetch((const void*)&xv[i + stride], 0, 0);

        uint32_t m   = (uint32_t)(i & (VEC_PER_ROW - 1)) << 2; // element index mod 8192
        uint32_t t   = m & ~(m << 1) & 0x1554u;                // pairs equal to '10'
        uint32_t s01 = (uint32_t)(__popc(t) & 1) << 31;        // sign for lanes 0,1
        uint32_t s23 = (uint32_t)(__popc(t & ~4u) & 1) << 31;  // lanes 2,3

        v4u v = __builtin_nontemporal_load(&xv[i]);
        v.x ^= s01; v.y ^= s01; v.z ^= s23; v.w ^= s23;
        __builtin_nontemporal_store(v, &ov[i]);
    }
}

extern "C" void kernel_launch(void* const* d_in, const int* in_sizes, int n_in,
                              void* d_out, int out_size, void* d_ws, size_t ws_size,
                              hipStream_t stream) {
    const float* x    = (const float*)d_in[0];   // [4096, 2, 8192] fp32
    const float* tmat = (const float*)d_in[1];   // [8192, 8192] fp32 (diagonal +-1)
    float*       out  = (float*)d_out;           // [4096, 2, 8192] fp32

    const long total_vec = (long)out_size / 4;   // 2^24 float4s

    const int block = 256;                       // 8 wave32 per block

    const bool row_exact = (total_vec & (VEC_PER_ROW - 1)) == 0;
    if (row_exact && ws_size >= (size_t)NSTATES * sizeof(float)) {
        float* diag = (float*)d_ws;
        cphase_extract_diag<<<(NSTATES + 255) / 256, 256, 0, stream>>>(tmat, diag);

        const long nrows = total_vec >> VEC_ROW_LOG2;          // 8192 rows
        // nthreads must be a multiple of VEC_PER_ROW; 16384*256 = 2^22 (4 rows/thread)
        long want_blocks = (nrows * VEC_PER_ROW + block - 1) / block;
        int  grid = (int)(want_blocks < 16384 ? want_blocks : 16384);
        // round grid so grid*block is a multiple of VEC_PER_ROW (block=256 -> grid%8==0)
        grid = (grid + 7) & ~7;
        cphase_diag_scale<<<grid, block, 0, stream>>>(x, diag, out, nrows);
    } else {
        long want_grid = (total_vec + block - 1) / block;
        int  grid = (int)(want_grid < 16384 ? want_grid : 16384);
        cphase_sign_flip<<<grid, block, 0, stream>>>(x, out, total_vec);
    }
}